// PainnDensityModel_50242527428642
// MI455X (gfx1250) — compile-verified
//
#include <hip/hip_runtime.h>
#include <math.h>

// Problem constants (must match reference)
#define NA      2000      // atoms
#define NP      10000     // probes
#define EA_CNT  60000     // atom edges
#define EP_CNT  150000    // probe edges
#define F       128
#define KB      30        // sinc basis size
#define KSP     32        // padded basis stride (multiple of 4 -> no K guards)
#define LAYERS  3
#define CUT     4.0f
#define CH      10000     // edge chunk: divides EA (6) and EP (15); multiple of 16
#define PI_F    3.14159265358979323846f

typedef __attribute__((ext_vector_type(2))) float v2f;
typedef __attribute__((ext_vector_type(8))) float v8f;

__device__ __forceinline__ float silu_f(float x)  { return x / (1.0f + expf(-x)); }
__device__ __forceinline__ float sigm_f(float x)  { return 1.0f / (1.0f + expf(-x)); }

// ---------------------------------------------------------------------------
// Generic fp32 WMMA GEMM:  C[M x Nc] = act(A[M x Kd] @ B[Kd x Nc] + bias)
// PRECONDITIONS (enforced by caller): M % 16 == 0, Nc % 16 == 0, Kd % 4 == 0.
// -> no bounds guards, no EXEC juggling; inner loop is pure load + WMMA.
// One wave32 computes one 16x16 tile via V_WMMA_F32_16X16X4_F32, stepping K by 4.
// A fragment (16x4 f32): lane&15 = m; lane>>4 selects K-pair; regs hold (k, k+1).
// B fragment (4x16 f32): lane&15 = n; lanes 0-15 hold K-pair {0,1}, 16-31 {2,3}.
// C/D (16x16 f32): VGPR r holds M = r + 8*(lane>>4), N = lane&15.
// ACT: 0 = none, 1 = silu, 2 = sigmoid
// ---------------------------------------------------------------------------
template<int ACT>
__global__ void gemm_wmma_kernel(const float* __restrict__ A,
                                 const float* __restrict__ B,
                                 const float* __restrict__ bias,
                                 float* __restrict__ C,
                                 int M, int Kd, int Nc)
{
    const int tiles_n = Nc >> 4;
    const int tiles_m = M >> 4;
    const int tile = blockIdx.x * (blockDim.x >> 5) + (threadIdx.x >> 5);
    if (tile >= tiles_m * tiles_n) return;         // wave-uniform branch
    const int tm = tile / tiles_n;
    const int tn = tile - tm * tiles_n;

    const int lane = threadIdx.x & 31;
    const int half = lane >> 4;                    // 0 or 1
    const int i16  = lane & 15;
    const int rowA = tm * 16 + i16;                // row this lane loads from A
    const int colB = tn * 16 + i16;                // col this lane loads from B

    const float* __restrict__ pA = A + (size_t)rowA * Kd + 2 * half; // 8B aligned
    const float* __restrict__ pB = B + (size_t)(2 * half) * Nc + colB;

    v8f acc = {0.f, 0.f, 0.f, 0.f, 0.f, 0.f, 0.f, 0.f};

#pragma unroll 4
    for (int k0 = 0; k0 < Kd; k0 += 4) {
        const v2f a = *(const v2f*)(pA + k0);      // one global_load_b64
        v2f b;
        b.x = pB[(size_t)k0 * Nc];
        b.y = pB[(size_t)(k0 + 1) * Nc];
        acc = __builtin_amdgcn_wmma_f32_16x16x4_f32(false, a, false, b,
                                                    (short)0, acc, false, false);
    }

    const float bn = bias ? bias[colB] : 0.0f;
    float* __restrict__ pC = C + (size_t)(tm * 16 + 8 * half) * Nc + colB;
#pragma unroll
    for (int r = 0; r < 8; ++r) {
        float x = acc[r] + bn;
        if (ACT == 1)      x = silu_f(x);
        else if (ACT == 2) x = sigm_f(x);
        pC[(size_t)r * Nc] = x;
    }
}

// ---------------------------------------------------------------------------
// Edge geometry: diff, dist, unit vec, cosine cutoff, sinc expansion
// (state is written with row stride KSP=32; cols 30,31 zero-padded)
// ---------------------------------------------------------------------------
__global__ void edge_geom_kernel(const float* __restrict__ src_xyz,
                                 const float* __restrict__ dst_xyz,
                                 const int*   __restrict__ edges,
                                 const float* __restrict__ disp,
                                 const float* __restrict__ cell,
                                 float* __restrict__ unit,
                                 float* __restrict__ dist,
                                 float* __restrict__ cc,
                                 float* __restrict__ state, int E)
{
    const int e = blockIdx.x * blockDim.x + threadIdx.x;
    if (e >= E) return;
    const int s = edges[2 * e], d = edges[2 * e + 1];
    float dif[3];
#pragma unroll
    for (int j = 0; j < 3; ++j) {
        const float dc = disp[3 * e + 0] * cell[0 * 3 + j]
                       + disp[3 * e + 1] * cell[1 * 3 + j]
                       + disp[3 * e + 2] * cell[2 * 3 + j];
        dif[j] = dst_xyz[3 * d + j] - (src_xyz[3 * s + j] + dc);
    }
    const float dd = sqrtf(dif[0] * dif[0] + dif[1] * dif[1] + dif[2] * dif[2]);
    const float inv = 1.0f / fmaxf(dd, 1e-12f);
    unit[3 * e + 0] = dif[0] * inv;
    unit[3 * e + 1] = dif[1] * inv;
    unit[3 * e + 2] = dif[2] * inv;
    dist[e] = dd;
    cc[e] = (dd < CUT) ? 0.5f * (cosf(PI_F * dd / CUT) + 1.0f) : 0.0f;
#pragma unroll 5
    for (int k = 0; k < KB; ++k) {
        const float kk = (float)(k + 1);
        // sinc(d*k/c)*(pi*k/c) == sin(pi*k*d/c)/d  (limit pi*k/c at d->0)
        state[(size_t)e * KSP + k] =
            (dd > 1e-12f) ? sinf(PI_F * kk * dd / CUT) / dd : PI_F * kk / CUT;
    }
    state[(size_t)e * KSP + 30] = 0.0f;            // K padding
    state[(size_t)e * KSP + 31] = 0.0f;
}

__global__ void gather_embed_kernel(const float* __restrict__ emb,
                                    const int* __restrict__ Z,
                                    float* __restrict__ s, int n)
{
    const int t = blockIdx.x * blockDim.x + threadIdx.x;
    if (t >= n * F) return;
    const int node = t >> 7, f = t & (F - 1);
    s[t] = emb[(size_t)Z[node] * F + f];
}

__global__ void zero_kernel(float* __restrict__ p, int n)
{
    const int t = blockIdx.x * blockDim.x + threadIdx.x;
    if (t < n) p[t] = 0.0f;
}

__global__ void copy_kernel(float* __restrict__ dst, const float* __restrict__ src, int n)
{
    const int t = blockIdx.x * blockDim.x + threadIdx.x;
    if (t < n) dst[t] = src[t];
}

// ---------------------------------------------------------------------------
// Edge message + scatter-add (segment_sum via hardware fp32 atomics).
// filt holds (state@fw + fb) for this chunk; cc applied here.
//   out[e,j] = filt[e,j] * cc[e] * phi[src,j] ; j in [0,3F)
//   g_sv=out[:,0:F], g_ev=out[:,F:2F], g_ss=out[:,2F:3F]
//   s_acc[dst,f]   += g_ss
//   v_acc[dst,i,f] += v_send[src,i,f]*g_sv + g_ev*unit[e,i]
// ---------------------------------------------------------------------------
__global__ void edge_scatter_kernel(const int* __restrict__ edges, int e0, int nE,
                                    const float* __restrict__ filt,
                                    const float* __restrict__ cc,
                                    const float* __restrict__ unit,
                                    const float* __restrict__ phi,
                                    const float* __restrict__ v_send,
                                    float* __restrict__ s_acc,
                                    float* __restrict__ v_acc)
{
    const int t = blockIdx.x * blockDim.x + threadIdx.x;
    if (t >= nE * F) return;
    const int el = t >> 7, f = t & (F - 1);
    const int e = e0 + el;
    const int src = edges[2 * e], dst = edges[2 * e + 1];
    const float c = cc[e];
    const float* fr = filt + (size_t)el * (3 * F);
    const float* pr = phi + (size_t)src * (3 * F);
    const float g_sv = fr[f]         * c * pr[f];
    const float g_ev = fr[F + f]     * c * pr[F + f];
    const float g_ss = fr[2 * F + f] * c * pr[2 * F + f];
    unsafeAtomicAdd(&s_acc[(size_t)dst * F + f], g_ss);
#pragma unroll
    for (int i = 0; i < 3; ++i) {
        const float m = v_send[((size_t)src * 3 + i) * F + f] * g_sv
                      + g_ev * unit[3 * e + i];
        unsafeAtomicAdd(&v_acc[((size_t)dst * 3 + i) * F + f], m);
    }
}

// cat = [s, ||Vv||_axis1]  (n x 2F)
__global__ void painn_prep_kernel(const float* __restrict__ s,
                                  const float* __restrict__ Vv,
                                  float* __restrict__ cat, int n)
{
    const int t = blockIdx.x * blockDim.x + threadIdx.x;
    if (t >= n * F) return;
    const int node = t >> 7, f = t & (F - 1);
    const float* vr = Vv + (size_t)node * 3 * F;
    const float a0 = vr[f], a1 = vr[F + f], a2 = vr[2 * F + f];
    cat[(size_t)node * 2 * F + f]     = s[t];
    cat[(size_t)node * 2 * F + F + f] = sqrtf(a0 * a0 + a1 * a1 + a2 * a2);
}

// s += a_ss + a_sv*inner ; v += a_vv * Uv
__global__ void painn_apply_kernel(const float* __restrict__ a,
                                   const float* __restrict__ Uv,
                                   const float* __restrict__ Vv,
                                   float* __restrict__ s,
                                   float* __restrict__ v, int n)
{
    const int t = blockIdx.x * blockDim.x + threadIdx.x;
    if (t >= n * F) return;
    const int node = t >> 7, f = t & (F - 1);
    const float* ur = Uv + (size_t)node * 3 * F;
    const float* vr = Vv + (size_t)node * 3 * F;
    const float inner = ur[f] * vr[f] + ur[F + f] * vr[F + f]
                      + ur[2 * F + f] * vr[2 * F + f];
    const float* ar = a + (size_t)node * 3 * F;
    s[t] += ar[f] + ar[F + f] * inner;
    const float avv = ar[2 * F + f];
    float* vd = v + (size_t)node * 3 * F;
    vd[f]         += avv * ur[f];
    vd[F + f]     += avv * ur[F + f];
    vd[2 * F + f] += avv * ur[2 * F + f];
}

// ps = ps*gs + (1-gs)*m_s ; pv = pv*gv + (1-gv)*m_v
__global__ void probe_blend_kernel(const float* __restrict__ gate,
                                   const float* __restrict__ ms,
                                   const float* __restrict__ mv,
                                   float* __restrict__ ps,
                                   float* __restrict__ pv, int n)
{
    const int t = blockIdx.x * blockDim.x + threadIdx.x;
    if (t >= n * F) return;
    const int p = t >> 7, f = t & (F - 1);
    const float gs = gate[(size_t)p * 2 * F + f];
    const float gv = gate[(size_t)p * 2 * F + F + f];
    ps[t] = ps[t] * gs + (1.0f - gs) * ms[t];
#pragma unroll
    for (int i = 0; i < 3; ++i) {
        const size_t idx = ((size_t)p * 3 + i) * F + f;
        pv[idx] = pv[idx] * gv + (1.0f - gv) * mv[idx];
    }
}

// ---------------------------------------------------------------------------
extern "C" void kernel_launch(void* const* d_in, const int* in_sizes, int n_in,
                              void* d_out, int out_size, void* d_ws, size_t ws_size,
                              hipStream_t stream)
{
    const float* atom_xyz  = (const float*)d_in[0];
    const float* probe_xyz = (const float*)d_in[1];
    const float* cell      = (const float*)d_in[2];
    const float* a_disp    = (const float*)d_in[3];
    const float* p_disp    = (const float*)d_in[4];
    const float* atom_emb  = (const float*)d_in[5];
    const float* ai_fw = (const float*)d_in[6];   const float* ai_fb = (const float*)d_in[7];
    const float* ai_w1 = (const float*)d_in[8];   const float* ai_b1 = (const float*)d_in[9];
    const float* ai_w2 = (const float*)d_in[10];  const float* ai_b2 = (const float*)d_in[11];
    const float* au_U  = (const float*)d_in[12];  const float* au_V  = (const float*)d_in[13];
    const float* au_w1 = (const float*)d_in[14];  const float* au_b1 = (const float*)d_in[15];
    const float* au_w2 = (const float*)d_in[16];  const float* au_b2 = (const float*)d_in[17];
    const float* pm_fw = (const float*)d_in[18];  const float* pm_fb = (const float*)d_in[19];
    const float* pm_w1 = (const float*)d_in[20];  const float* pm_b1 = (const float*)d_in[21];
    const float* pm_w2 = (const float*)d_in[22];  const float* pm_b2 = (const float*)d_in[23];
    const float* pm_gw1= (const float*)d_in[24];  const float* pm_gb1= (const float*)d_in[25];
    const float* pm_gw2= (const float*)d_in[26];  const float* pm_gb2= (const float*)d_in[27];
    const float* pu_U  = (const float*)d_in[28];  const float* pu_V  = (const float*)d_in[29];
    const float* pu_w1 = (const float*)d_in[30];  const float* pu_b1 = (const float*)d_in[31];
    const float* pu_w2 = (const float*)d_in[32];  const float* pu_b2 = (const float*)d_in[33];
    const int* nodes_Z     = (const int*)d_in[34];
    const int* atom_edges  = (const int*)d_in[35];
    const int* probe_edges = (const int*)d_in[36];

    // ---- workspace layout (floats) -------------------------------------
    float* W = (float*)d_ws;
    size_t off = 0;
    auto alloc = [&](size_t n) { float* p = W + off; off += n; return p; };
    float* a_unit  = alloc((size_t)EA_CNT * 3);
    float* a_dist  = alloc(EA_CNT);
    float* a_cc    = alloc(EA_CNT);
    float* a_state = alloc((size_t)EA_CNT * KSP);   // padded to 32 cols
    float* p_unit  = alloc((size_t)EP_CNT * 3);
    float* p_dist  = alloc(EP_CNT);
    float* p_cc    = alloc(EP_CNT);
    float* p_state = alloc((size_t)EP_CNT * KSP);   // padded to 32 cols
    float* s_buf   = alloc((size_t)NA * F);
    float* v_buf   = alloc((size_t)NA * 3 * F);
    float* v_old   = alloc((size_t)NA * 3 * F);
    float* s_list  = alloc((size_t)LAYERS * NA * F);
    float* v_list  = alloc((size_t)LAYERS * NA * 3 * F);
    float* phi     = alloc((size_t)NA * 3 * F);
    float* filt    = alloc((size_t)CH * 3 * F);
    float* m_s     = alloc((size_t)NP * F);       // m_s / m_v contiguous -> one zero
    float* m_v     = alloc((size_t)NP * 3 * F);
    float* Uv      = alloc((size_t)NP * 3 * F);
    float* Vv      = alloc((size_t)NP * 3 * F);
    float* catb    = alloc((size_t)NP * 2 * F);
    float* hbuf    = alloc((size_t)NP * F);
    float* abuf    = alloc((size_t)NP * 3 * F);
    float* g2      = alloc((size_t)NP * 2 * F);
    if (off * sizeof(float) > ws_size) return;    // insufficient scratch -> bail

    const int T = 256;
    auto gb = [](int n) { return (n + 255) / 256; };
    // M % 16 == 0, Kd % 4 == 0, Nc % 16 == 0 required.
    auto gemm = [&](const float* A, const float* B, const float* bias, float* C,
                    int M, int Kd, int Nc, int act) {
        const int tiles = (M / 16) * (Nc / 16);
        const int blocks = (tiles + 3) / 4;      // 4 waves / block, 1 tile / wave
        if (act == 1)
            gemm_wmma_kernel<1><<<blocks, 128, 0, stream>>>(A, B, bias, C, M, Kd, Nc);
        else if (act == 2)
            gemm_wmma_kernel<2><<<blocks, 128, 0, stream>>>(A, B, bias, C, M, Kd, Nc);
        else
            gemm_wmma_kernel<0><<<blocks, 128, 0, stream>>>(A, B, bias, C, M, Kd, Nc);
    };
    // filter GEMM: A = padded sinc state (row stride KSP), weights are KB x 3F
    // rows; since state cols 30,31 are zero, we can treat B as if it had 32 rows
    // ONLY if we never read rows 30,31 of B -> they ARE read (k=30,31 of B).
    // So B must also provide 32 rows: we read fw rows 0..29 normally and rows
    // 30,31 would be OOB. Instead: zero-padded state means a.x/a.y are 0 for
    // k>=30, so the product contribution is 0 regardless of B values -- but the
    // LOAD of B rows 30,31 would still be OOB. Avoid by staging fw into a
    // padded 32 x 3F scratch copy per layer.
    float* fw_pad = alloc((size_t)KSP * 3 * F);
    if (off * sizeof(float) > ws_size) return;

    // ---- edge featurization --------------------------------------------
    edge_geom_kernel<<<gb(EA_CNT), T, 0, stream>>>(atom_xyz, atom_xyz, atom_edges,
        a_disp, cell, a_unit, a_dist, a_cc, a_state, EA_CNT);
    edge_geom_kernel<<<gb(EP_CNT), T, 0, stream>>>(atom_xyz, probe_xyz, probe_edges,
        p_disp, cell, p_unit, p_dist, p_cc, p_state, EP_CNT);

    // ---- atom representation init --------------------------------------
    gather_embed_kernel<<<gb(NA * F), T, 0, stream>>>(atom_emb, nodes_Z, s_buf, NA);
    zero_kernel<<<gb(NA * 3 * F), T, 0, stream>>>(v_buf, NA * 3 * F);

    // ---- atom message-passing layers -----------------------------------
    for (int l = 0; l < LAYERS; ++l) {
        // phi = mlp2(s)  : N x 3F
        gemm(s_buf, ai_w1 + (size_t)l * F * F, ai_b1 + (size_t)l * F, hbuf, NA, F, F, 1);
        gemm(hbuf, ai_w2 + (size_t)l * F * 3 * F, ai_b2 + (size_t)l * 3 * F, phi, NA, F, 3 * F, 0);
        // snapshot v so messages read the pre-update state
        copy_kernel<<<gb(NA * 3 * F), T, 0, stream>>>(v_old, v_buf, NA * 3 * F);
        // stage filter weights into 32-row padded buffer (rows 30,31 zeroed)
        zero_kernel<<<gb(KSP * 3 * F), T, 0, stream>>>(fw_pad, KSP * 3 * F);
        copy_kernel<<<gb(KB * 3 * F), T, 0, stream>>>(fw_pad,
            ai_fw + (size_t)l * KB * 3 * F, KB * 3 * F);
        for (int c = 0; c < EA_CNT / CH; ++c) {
            const int e0 = c * CH;
            gemm(a_state + (size_t)e0 * KSP, fw_pad, ai_fb + (size_t)l * 3 * F,
                 filt, CH, KSP, 3 * F, 0);
            edge_scatter_kernel<<<gb(CH * F), T, 0, stream>>>(atom_edges, e0, CH,
                filt, a_cc, a_unit, phi, v_old, s_buf, v_buf);
        }
        // PaiNN update (atoms)
        gemm(v_buf, au_U + (size_t)l * F * F, nullptr, Uv, 3 * NA, F, F, 0);
        gemm(v_buf, au_V + (size_t)l * F * F, nullptr, Vv, 3 * NA, F, F, 0);
        painn_prep_kernel<<<gb(NA * F), T, 0, stream>>>(s_buf, Vv, catb, NA);
        gemm(catb, au_w1 + (size_t)l * 2 * F * F, au_b1 + (size_t)l * F, hbuf, NA, 2 * F, F, 1);
        gemm(hbuf, au_w2 + (size_t)l * F * 3 * F, au_b2 + (size_t)l * 3 * F, abuf, NA, F, 3 * F, 0);
        painn_apply_kernel<<<gb(NA * F), T, 0, stream>>>(abuf, Uv, Vv, s_buf, v_buf, NA);
        copy_kernel<<<gb(NA * F), T, 0, stream>>>(s_list + (size_t)l * NA * F, s_buf, NA * F);
        copy_kernel<<<gb(NA * 3 * F), T, 0, stream>>>(v_list + (size_t)l * NA * 3 * F, v_buf, NA * 3 * F);
    }

    // ---- probe phase ----------------------------------------------------
    float* ps = (float*)d_out;                      // P x F
    float* pv = (float*)d_out + (size_t)NP * F;     // P x 3 x F
    zero_kernel<<<gb(NP * 4 * F), T, 0, stream>>>(ps, NP * 4 * F);  // ps & pv contiguous

    for (int l = 0; l < LAYERS; ++l) {
        // phi = mlp2(s_list[l]) with probe-message weights
        gemm(s_list + (size_t)l * NA * F, pm_w1 + (size_t)l * F * F,
             pm_b1 + (size_t)l * F, hbuf, NA, F, F, 1);
        gemm(hbuf, pm_w2 + (size_t)l * F * 3 * F, pm_b2 + (size_t)l * 3 * F, phi, NA, F, 3 * F, 0);
        // zero message accumulators (m_s, m_v contiguous)
        zero_kernel<<<gb(NP * 4 * F), T, 0, stream>>>(m_s, NP * 4 * F);
        // stage probe filter weights (padded)
        zero_kernel<<<gb(KSP * 3 * F), T, 0, stream>>>(fw_pad, KSP * 3 * F);
        copy_kernel<<<gb(KB * 3 * F), T, 0, stream>>>(fw_pad,
            pm_fw + (size_t)l * KB * 3 * F, KB * 3 * F);
        for (int c = 0; c < EP_CNT / CH; ++c) {
            const int e0 = c * CH;
            gemm(p_state + (size_t)e0 * KSP, fw_pad, pm_fb + (size_t)l * 3 * F,
                 filt, CH, KSP, 3 * F, 0);
            edge_scatter_kernel<<<gb(CH * F), T, 0, stream>>>(probe_edges, e0, CH,
                filt, p_cc, p_unit, phi, v_list + (size_t)l * NA * 3 * F, m_s, m_v);
        }
        // gate = sigmoid(mlp2(m_s))
        gemm(m_s, pm_gw1 + (size_t)l * F * 2 * F, pm_gb1 + (size_t)l * 2 * F, catb, NP, F, 2 * F, 1);
        gemm(catb, pm_gw2 + (size_t)l * 2 * F * 2 * F, pm_gb2 + (size_t)l * 2 * F, g2, NP, 2 * F, 2 * F, 2);
        probe_blend_kernel<<<gb(NP * F), T, 0, stream>>>(g2, m_s, m_v, ps, pv, NP);
        // PaiNN update (probes)
        gemm(pv, pu_U + (size_t)l * F * F, nullptr, Uv, 3 * NP, F, F, 0);
        gemm(pv, pu_V + (size_t)l * F * F, nullptr, Vv, 3 * NP, F, F, 0);
        painn_prep_kernel<<<gb(NP * F), T, 0, stream>>>(ps, Vv, catb, NP);
        gemm(catb, pu_w1 + (size_t)l * 2 * F * F, pu_b1 + (size_t)l * F, hbuf, NP, 2 * F, F, 1);
        gemm(hbuf, pu_w2 + (size_t)l * F * 3 * F, pu_b2 + (size_t)l * 3 * F, abuf, NP, F, 3 * F, 0);
        painn_apply_kernel<<<gb(NP * F), T, 0, stream>>>(abuf, Uv, Vv, ps, pv, NP);
    }
}